// Torch_SINDy_54966991454902
// MI455X (gfx1250) — compile-verified
//
#include <hip/hip_runtime.h>

// ---------------------------------------------------------------------------
// SINDy feature regression: out[n] = poly(u)·c[0:3] + d·c[3:6] + d^T Cm poly(u)
// Memory-bound (~1.5 FLOP/byte): pure streaming on MI455X.
// Path: TDM (tensor_load_to_lds) double-buffered global->LDS pipeline,
//       NT b128 stores, scalar-cached coefficients.
// ---------------------------------------------------------------------------

typedef unsigned int v4u __attribute__((ext_vector_type(4)));
typedef int          v8i __attribute__((ext_vector_type(8)));
typedef int          v4i __attribute__((ext_vector_type(4)));
typedef float        v4f __attribute__((ext_vector_type(4)));

#define TILE_FLOATS   4096            // 16 KiB tile = 1024 rows of float4
#define ROWS_PER_TILE (TILE_FLOATS / 4)
#define BLOCK_THREADS 256             // 8 waves (wave32)
#define GRID_BLOCKS   2048

#if defined(__has_builtin)
#if __has_builtin(__builtin_amdgcn_tensor_load_to_lds) && __has_builtin(__builtin_amdgcn_s_wait_tensorcnt)
#define HAVE_TDM 1
#endif
#endif

// --- per-row math: column-factored bilinear form, qj = c[j] + (Cm^T d)[j] ---
__device__ __forceinline__ float row_eval(v4f r, const float* cc) {
  float u  = r.x, d1 = r.y, d2 = r.z, d3 = r.w;
  float u2 = u * u;
  float u3 = u2 * u;
  float q0 = __builtin_fmaf(d3, cc[12], __builtin_fmaf(d2, cc[9],  __builtin_fmaf(d1, cc[6],  cc[0])));
  float q1 = __builtin_fmaf(d3, cc[13], __builtin_fmaf(d2, cc[10], __builtin_fmaf(d1, cc[7],  cc[1])));
  float q2 = __builtin_fmaf(d3, cc[14], __builtin_fmaf(d2, cc[11], __builtin_fmaf(d1, cc[8],  cc[2])));
  return __builtin_fmaf(u,  q0,
         __builtin_fmaf(u2, q1,
         __builtin_fmaf(u3, q2,
         __builtin_fmaf(d1, cc[3],
         __builtin_fmaf(d2, cc[4], d3 * cc[5])))));
}

#ifdef HAVE_TDM
// Build D# (ISA cdna5 ch.8) for a 1-D tensor view and issue the TDM DMA.
// Group0: count=1, lds_addr, global_addr[56:0], type=2 ("image").
// Group1: data_size=4B, tensor_dim0=remaining floats (OOB reads -> 0),
//         tile_dim0=TILE_FLOATS, tile_dim1=1.
__device__ __forceinline__ void tdm_load_tile(const float* gsrc,
                                              unsigned lds_off,
                                              unsigned remaining_floats) {
  unsigned long long ga = (unsigned long long)gsrc;
  v4u g0;
  g0.x = 1u;                                         // count=1, user descriptor
  g0.y = lds_off;                                    // lds_addr (bytes)
  g0.z = (unsigned)ga;                               // global_addr[31:0]
  g0.w = (unsigned)((ga >> 32) & 0x01FFFFFFu)        // global_addr[56:32]
       | (2u << 30);                                 // type=2

  unsigned td0   = remaining_floats;                 // tensor_dim0
  unsigned td1   = 1u;                               // tensor_dim1
  unsigned tile0 = TILE_FLOATS;                      // tile_dim0 (16-bit field)
  unsigned long long st0 = remaining_floats;         // tensor_dim0_stride

  v8i g1;
  g1[0] = (int)(2u << 16);                           // wg_mask=0, data_size=2 (4B)
  g1[1] = (int)((td0 & 0xFFFFu) << 16);              // [63:48] tensor_dim0 lo16
  g1[2] = (int)((td0 >> 16) | ((td1 & 0xFFFFu) << 16));
  g1[3] = (int)((td1 >> 16) | (tile0 << 16));        // [127:112] tile_dim0
  g1[4] = (int)(1u);                                 // tile_dim1=1, tile_dim2=0
  g1[5] = (int)(unsigned)(st0 & 0xFFFFFFFFu);
  g1[6] = (int)(unsigned)((st0 >> 32) & 0xFFFFu);
  g1[7] = 0;

  v4i z4 = {0, 0, 0, 0};                             // groups 2/3 unused (<=2D)
  v8i z8 = {0, 0, 0, 0, 0, 0, 0, 0};                 // trailing group (6-arg form)
  __builtin_amdgcn_tensor_load_to_lds(g0, g1, z4, z4, z8, 0);
}
#endif

__global__ __launch_bounds__(BLOCK_THREADS)
void sindy_stream_kernel(const float* __restrict__ big_u,
                         const float* __restrict__ coeff,
                         const float* __restrict__ base,
                         float* __restrict__ out,
                         int nrows) {
  // c = coefficients * base_coeffs : uniform -> s_load + scalar-resident
  float cc[15];
#pragma unroll
  for (int i = 0; i < 15; ++i) cc[i] = coeff[i] * base[i];

  const int tid          = (int)threadIdx.x;
  const int ntiles       = (nrows + ROWS_PER_TILE - 1) / ROWS_PER_TILE;
  const int stride_tiles = (int)gridDim.x;

#ifdef HAVE_TDM
  // ---------------- TDM double-buffered pipeline ----------------
  __shared__ __align__(16) float tile[2][TILE_FLOATS];
  const unsigned total_floats = (unsigned)nrows * 4u;
  const bool wave0 = (tid >> 5) == 0;                 // one DMA issuer per WG
  int buf = 0;

  int t0i = (int)blockIdx.x;
  if (t0i < ntiles && wave0) {
    unsigned base_f = (unsigned)t0i * TILE_FLOATS;
    tdm_load_tile(big_u + base_f,
                  (unsigned)(unsigned long long)&tile[0][0],
                  total_floats - base_f);
  }

  for (int t = t0i; t < ntiles; t += stride_tiles) {
    int tn = t + stride_tiles;
    if (wave0) {
      if (tn < ntiles) {                              // DMA next tile now
        unsigned base_f = (unsigned)tn * TILE_FLOATS;
        tdm_load_tile(big_u + base_f,
                      (unsigned)(unsigned long long)&tile[buf ^ 1][0],
                      total_floats - base_f);
        __builtin_amdgcn_s_wait_tensorcnt((short)1);  // current tile complete
      } else {
        __builtin_amdgcn_s_wait_tensorcnt((short)0);
      }
    }
    __syncthreads();                                  // publish LDS tile

    // 4 consecutive rows per thread: ds_load_b128 x4, store b128 (NT)
    const int row0 = t * ROWS_PER_TILE + tid * 4;
    const v4f* lp = (const v4f*)&tile[buf][tid * 16];
    v4f o;
#pragma unroll
    for (int k = 0; k < 4; ++k) {
      o[k] = row_eval(lp[k], cc);
    }
    if (row0 + 3 < nrows) {
      __builtin_nontemporal_store(o, (v4f*)(out + row0));
    } else {
#pragma unroll
      for (int k = 0; k < 4; ++k)
        if (row0 + k < nrows) out[row0 + k] = o[k];
    }

    __syncthreads();                                  // tile fully consumed
    buf ^= 1;
  }
#else
  // ---------------- direct-load fallback (no TDM builtin) ----------------
  const int total_threads = stride_tiles * BLOCK_THREADS;
  for (int row0 = ((int)blockIdx.x * BLOCK_THREADS + tid) * 4;
       row0 < nrows; row0 += total_threads * 4) {
    v4f o;
#pragma unroll
    for (int k = 0; k < 4; ++k) {
      v4f r = __builtin_nontemporal_load((const v4f*)(big_u + (row0 + k) * 4));
      o[k] = row_eval(r, cc);
    }
    if (row0 + 3 < nrows) {
      __builtin_nontemporal_store(o, (v4f*)(out + row0));
    } else {
#pragma unroll
      for (int k = 0; k < 4; ++k)
        if (row0 + k < nrows) out[row0 + k] = o[k];
    }
  }
#endif
}

extern "C" void kernel_launch(void* const* d_in, const int* in_sizes, int n_in,
                              void* d_out, int out_size, void* d_ws, size_t ws_size,
                              hipStream_t stream) {
  const float* big_u = (const float*)d_in[0];   // (N,4) f32
  const float* coeff = (const float*)d_in[1];   // (15,) f32
  const float* base  = (const float*)d_in[2];   // (15,) f32
  float* out = (float*)d_out;                   // (N,1) f32

  const int nrows  = in_sizes[0] / 4;
  const int ntiles = (nrows + ROWS_PER_TILE - 1) / ROWS_PER_TILE;
  int grid = ntiles < GRID_BLOCKS ? ntiles : GRID_BLOCKS;
  if (grid < 1) grid = 1;

  sindy_stream_kernel<<<grid, BLOCK_THREADS, 0, stream>>>(big_u, coeff, base,
                                                          out, nrows);
}